// VQAMaskRCNNBenchmark_59596966199320
// MI455X (gfx1250) — compile-verified
//
#include <hip/hip_runtime.h>
#include <stdint.h>

// Problem constants (from reference)
#define B_IMG 16
#define N_BOX 512
#define N_CLS 1601
#define D_FEAT 2048
#define TOPK_K 100

typedef float    f32x2 __attribute__((ext_vector_type(2)));
typedef float    f32x8 __attribute__((ext_vector_type(8)));
typedef uint32_t u32x4 __attribute__((ext_vector_type(4)));
typedef int      i32x4 __attribute__((ext_vector_type(4)));
typedef int      i32x8 __attribute__((ext_vector_type(8)));

// ---------------------------------------------------------------------------
// wave32 shuffle helpers (explicit ds_bpermute; avoids any wave64 assumptions
// in headers since __AMDGCN_WAVEFRONT_SIZE is not defined for gfx1250)
// ---------------------------------------------------------------------------
__device__ __forceinline__ uint32_t lane_u32() { return threadIdx.x & 31u; }

__device__ __forceinline__ uint32_t shfl_u32(uint32_t v, uint32_t srcLane) {
  return (uint32_t)__builtin_amdgcn_ds_bpermute((int)(srcLane << 2), (int)v);
}
__device__ __forceinline__ float shfl_xor_f32(float v, uint32_t m) {
  return __uint_as_float(shfl_u32(__float_as_uint(v), lane_u32() ^ m));
}
__device__ __forceinline__ uint64_t shfl_xor_u64(uint64_t v, uint32_t m) {
  const uint32_t s  = lane_u32() ^ m;
  const uint32_t lo = shfl_u32((uint32_t)v, s);
  const uint32_t hi = shfl_u32((uint32_t)(v >> 32), s);
  return ((uint64_t)hi << 32) | lo;
}
// intra-wave LDS publish->consume ordering (wave32 lockstep + dscnt drain)
__device__ __forceinline__ void lds_fence_wave() {
  __builtin_amdgcn_wave_barrier();
  asm volatile("s_wait_dscnt 0" ::: "memory");
  __builtin_amdgcn_wave_barrier();
}

// ---------------------------------------------------------------------------
// Kernel A: per-box logsumexp over 1601 classes. One wave per box row.
// Final 32-lane sum uses V_WMMA_F32_16X16X4_F32 (exact f32 reduction):
//   A(16x4): VGPR0 = per-lane partial (K=0 rows 0..15, K=2 rows 0..15 from
//   lanes 16..31), VGPR1 = 0. B = all-ones -> D[m][n] = p[m] + p[m+16].
//   Rows 0..7 live in D v0..v7 on lanes 0..15, rows 8..15 on lanes 16..31,
//   so sum(v0..v7) + shfl_xor(,16) = full 32-lane sum.
// ---------------------------------------------------------------------------
__global__ void __launch_bounds__(256) logz_kernel(const float* __restrict__ scores,
                                                   float* __restrict__ logZ) {
  const int row      = blockIdx.x * 8 + (threadIdx.x >> 5); // 0..8191 exact
  const uint32_t l   = lane_u32();
  const float* p     = scores + (size_t)row * N_CLS;

  float m = -3.4e38f;
  for (int c = (int)l; c < N_CLS; c += 32) m = fmaxf(m, p[c]);
  for (int j = 16; j >= 1; j >>= 1) m = fmaxf(m, shfl_xor_f32(m, (uint32_t)j));

  float s = 0.0f;
  for (int c = (int)l; c < N_CLS; c += 32) s += expf(p[c] - m);

  f32x2 Av; Av.x = s;    Av.y = 0.0f;    // K=1,3 contribute zero
  f32x2 Bv; Bv.x = 1.0f; Bv.y = 1.0f;    // all-ones B: layout-independent
  f32x8 Cv = {0.f, 0.f, 0.f, 0.f, 0.f, 0.f, 0.f, 0.f};
  f32x8 Dv = __builtin_amdgcn_wmma_f32_16x16x4_f32(
      /*neg_a=*/false, Av, /*neg_b=*/false, Bv,
      /*c_mod=*/(short)0, Cv, /*reuse_a=*/false, /*reuse_b=*/false);
  float s8  = Dv[0] + Dv[1] + Dv[2] + Dv[3] + Dv[4] + Dv[5] + Dv[6] + Dv[7];
  float tot = s8 + shfl_xor_f32(s8, 16);

  if (l == 0) logZ[row] = m + logf(tot);
}

// ---------------------------------------------------------------------------
// Kernel B: dets = boxes/scale, areas, zero max_conf. Per-image 10KB tile in
// ws laid out [512 x float4 dets][512 x float area] so TDM can fetch it 1D.
// ---------------------------------------------------------------------------
__global__ void __launch_bounds__(512) prep_kernel(const float* __restrict__ boxes,
                                                   const float* __restrict__ im_scales,
                                                   float* __restrict__ tiles,
                                                   float* __restrict__ max_conf) {
  const int b = blockIdx.x, n = threadIdx.x;
  const float4 bx = ((const float4*)boxes)[b * N_BOX + n];
  const float sc = im_scales[b];
  const float x1 = bx.x / sc, y1 = bx.y / sc, x2 = bx.z / sc, y2 = bx.w / sc;
  float* tile = tiles + (size_t)b * 2560;
  ((float4*)tile)[n] = make_float4(x1, y1, x2, y2);
  tile[2048 + n]     = (x2 - x1) * (y2 - y1);
  max_conf[b * N_BOX + n] = 0.0f;
}

// ---------------------------------------------------------------------------
// In-register bitonic sort of 512 u64 keys per wave (16 regs/lane).
// Element e = r*32 + lane. j>=32 -> in-lane register CX; j<32 -> shfl_xor CX.
// ---------------------------------------------------------------------------
template <int JR>
__device__ __forceinline__ void cx_reg(uint64_t key[16], unsigned k) {
#pragma unroll
  for (int r = 0; r < 16; ++r) {
    if ((r & JR) == 0) {
      const int r2 = r + JR;                       // compile-time after unroll
      const bool up = (((unsigned)(r << 5) & k) == 0);
      const uint64_t a = key[r], b = key[r2];
      const uint64_t mn = a < b ? a : b;
      const uint64_t mx = a < b ? b : a;
      key[r]  = up ? mn : mx;
      key[r2] = up ? mx : mn;
    }
  }
}

__device__ __forceinline__ void cx_lane(uint64_t key[16], unsigned j, unsigned k,
                                        unsigned l) {
#pragma unroll
  for (int r = 0; r < 16; ++r) {
    const uint64_t o = shfl_xor_u64(key[r], j);
    const unsigned e = (unsigned)(r << 5) + l;
    const bool up  = ((e & k) == 0);
    const bool low = ((l & j) == 0);
    const uint64_t mn = key[r] < o ? key[r] : o;
    const uint64_t mx = key[r] < o ? o : key[r];
    key[r] = (low == up) ? mn : mx;
  }
}

// ---------------------------------------------------------------------------
// Kernel C: NMS. One wave32 per (image, class); 8 classes per 256-thread block
// (consecutive classes share scores cachelines). Boxes tile arrives in LDS via
// the Tensor Data Mover. Sorted order is register-resident; keep flags are a
// 16-bit mask per lane; keep[i] broadcast via shfl; suppression test is the
// division-free iou>0.5 <=> 3*inter > area_i + area_e.
// ---------------------------------------------------------------------------
__global__ void __launch_bounds__(256) nms_kernel(const float* __restrict__ scores,
                                                  const float* __restrict__ logZ,
                                                  const float* __restrict__ tiles,
                                                  float* __restrict__ max_conf) {
  __shared__ __align__(16) float s_tile[2560];   // dets(2048) + areas(512)
  __shared__ uint32_t s_ord[8][N_BOX];           // per-wave sorted box indices

  const int b   = blockIdx.x / 200;
  const int grp = blockIdx.x % 200;
  const int w   = threadIdx.x >> 5;
  const uint32_t l = lane_u32();
  const int cls = 1 + grp * 8 + w;               // classes 1..1600 (skip bg)

  // --- TDM: async 1D tile (2560 f32 = 10KB) global -> LDS; per-wave issue ---
  {
    const uint64_t ga  = (uint64_t)(uintptr_t)(tiles + (size_t)b * 2560);
    const uint32_t lds = (uint32_t)(uintptr_t)(&s_tile[0]); // LDS byte offset
    u32x4 g0;
    g0.x = 1u;                                       // count=1, user mode
    g0.y = lds;                                      // lds_addr
    g0.z = (uint32_t)ga;                             // global_addr[31:0]
    g0.w = (uint32_t)((ga >> 32) & 0x01FFFFFFu) | 0x80000000u; // ga[56:32]|type=2
    i32x8 g1;
    g1[0] = (int)(2u << 16);                         // data_size = 4B
    g1[1] = (int)((2560u & 0xFFFFu) << 16);          // tensor_dim0 lo16
    g1[2] = (int)(((2560u >> 16) & 0xFFFFu) | (1u << 16)); // td0 hi | td1 lo=1
    g1[3] = (int)(2560u << 16);                      // td1 hi=0 | tile_dim0=2560
    g1[4] = 1;                                       // tile_dim1=1, tile_dim2=0
    g1[5] = 2560;                                    // tensor_dim0_stride lo32
    g1[6] = 0;
    g1[7] = 0;
    i32x4 z4 = {0, 0, 0, 0};                         // groups 2/3 unused (<=2D)
    i32x8 z8 = {0, 0, 0, 0, 0, 0, 0, 0};             // extra group (clang-23 form)
    __builtin_amdgcn_tensor_load_to_lds(g0, g1, z4, z4, z8, 0);
    __builtin_amdgcn_s_wait_tensorcnt(0);
    asm volatile("" ::: "memory");                   // TDM wrote LDS: no undef-fold
  }

  // --- per-class probabilities + sort keys -------------------------------
  // key = (~bits(prob) << 32) | n : ascending sort => prob desc, ties n asc
  uint64_t key[16];
#pragma unroll
  for (int r = 0; r < 16; ++r) {
    const int n = (r << 5) + (int)l;
    const float sc = scores[((size_t)b * N_BOX + n) * N_CLS + cls];
    const float pr = expf(sc - logZ[b * N_BOX + n]);
    key[r] = ((uint64_t)(~__float_as_uint(pr)) << 32) | (uint32_t)n;
  }

  for (unsigned k = 2; k <= 512; k <<= 1) {
    for (unsigned j = k >> 1; j; j >>= 1) {
      if (j >= 32) {
        if      (j == 256) cx_reg<8>(key, k);
        else if (j == 128) cx_reg<4>(key, k);
        else if (j == 64)  cx_reg<2>(key, k);
        else               cx_reg<1>(key, k);
      } else {
        cx_lane(key, j, k, l);
      }
    }
  }

  // --- publish sorted order; preload my elements' geometry ----------------
  uint32_t o[16], pb[16];
  float bx1[16], by1[16], bx2[16], by2[16], bar[16];
#pragma unroll
  for (int r = 0; r < 16; ++r) {
    o[r]  = (uint32_t)key[r];
    pb[r] = ~(uint32_t)(key[r] >> 32);               // prob bits (positive f32)
    s_ord[w][(r << 5) + (int)l] = o[r];
    const float4 bb = ((const float4*)s_tile)[o[r]];
    bx1[r] = bb.x; by1[r] = bb.y; bx2[r] = bb.z; by2[r] = bb.w;
    bar[r] = s_tile[2048 + o[r]];
  }
  lds_fence_wave();

  // --- greedy suppression scan -------------------------------------------
  uint32_t keep = 0xFFFFu;                           // bit r <-> element r*32+l
  for (int i = 0; i < N_BOX - 1; ++i) {
    const uint32_t km = shfl_u32(keep, (uint32_t)(i & 31));
    if (((km >> (i >> 5)) & 1u) == 0u) continue;     // i was suppressed (uniform)
    const uint32_t oi = s_ord[w][i];                 // uniform LDS broadcast
    const float4 bi = ((const float4*)s_tile)[oi];
    const float  ai = s_tile[2048 + oi];
#pragma unroll
    for (int r = 0; r < 16; ++r) {
      const unsigned e = (unsigned)(r << 5) + l;
      if (((keep >> r) & 1u) && e > (unsigned)i) {
        const float ix1 = fmaxf(bi.x, bx1[r]);
        const float iy1 = fmaxf(bi.y, by1[r]);
        const float ix2 = fminf(bi.z, bx2[r]);
        const float iy2 = fminf(bi.w, by2[r]);
        const float inter = fmaxf(ix2 - ix1, 0.0f) * fmaxf(iy2 - iy1, 0.0f);
        if (3.0f * inter > ai + bar[r]) keep &= ~(1u << r);   // iou > 0.5
      }
    }
  }

  // --- fold kept probs into per-box max_conf (order-independent) ----------
#pragma unroll
  for (int r = 0; r < 16; ++r)
    if ((keep >> r) & 1u)
      atomicMax((unsigned int*)(max_conf + b * N_BOX + o[r]), pb[r]);
}

// ---------------------------------------------------------------------------
// Kernel D1: top-100 boxes per image by max_conf (LDS bitonic, stable ties).
// ---------------------------------------------------------------------------
__global__ void __launch_bounds__(512) topk_kernel(const float* __restrict__ max_conf,
                                                   int* __restrict__ topk_idx) {
  __shared__ uint64_t sk[N_BOX];
  const int b = blockIdx.x, t = threadIdx.x;
  const float c = max_conf[b * N_BOX + t];
  sk[t] = ((uint64_t)(~__float_as_uint(c)) << 32) | (uint32_t)t; // asc => conf desc
  for (unsigned k = 2; k <= 512; k <<= 1)
    for (unsigned j = k >> 1; j; j >>= 1) {
      __syncthreads();
      const unsigned p = (unsigned)t ^ j;
      if (p > (unsigned)t) {
        const uint64_t a = sk[t], bb = sk[p];
        const bool up = (((unsigned)t & k) == 0);
        if ((a > bb) == up) { sk[t] = bb; sk[p] = a; }
      }
    }
  __syncthreads();
  if (t < TOPK_K) topk_idx[b * 128 + t] = (int)(sk[t] & 0xFFFFFFFFu);
}

// ---------------------------------------------------------------------------
// Kernel D2: gather features of kept boxes. 13 MB coalesced float4 copy.
// ---------------------------------------------------------------------------
__global__ void __launch_bounds__(256) gather_kernel(const float* __restrict__ features,
                                                     const int* __restrict__ topk_idx,
                                                     float* __restrict__ out) {
  const int b = blockIdx.x / TOPK_K, k = blockIdx.x % TOPK_K;
  const int idx = topk_idx[b * 128 + k];
  const float4* src = (const float4*)(features + ((size_t)b * N_BOX + idx) * D_FEAT);
  float4*       dst = (float4*)(out + ((size_t)b * TOPK_K + k) * D_FEAT);
  for (int t = threadIdx.x; t < D_FEAT / 4; t += 256) dst[t] = src[t];
}

// ---------------------------------------------------------------------------
// Launch. ws layout (floats): [0,40960) per-image tiles; [40960,49152) logZ;
// [49152,57344) max_conf; then B*128 ints of top-k indices. ~232 KB total.
// ---------------------------------------------------------------------------
extern "C" void kernel_launch(void* const* d_in, const int* in_sizes, int n_in,
                              void* d_out, int out_size, void* d_ws, size_t ws_size,
                              hipStream_t stream) {
  const float* boxes     = (const float*)d_in[0];
  const float* scores    = (const float*)d_in[1];
  const float* features  = (const float*)d_in[2];
  const float* im_scales = (const float*)d_in[3];
  float* out = (float*)d_out;

  float* ws_f    = (float*)d_ws;
  float* tiles   = ws_f;                  // 16 * 2560
  float* logZ    = ws_f + 16 * 2560;      // 8192
  float* maxconf = logZ + 8192;           // 8192
  int*   topk    = (int*)(maxconf + 8192);

  prep_kernel  <<<B_IMG, 512, 0, stream>>>(boxes, im_scales, tiles, maxconf);
  logz_kernel  <<<(B_IMG * N_BOX) / 8, 256, 0, stream>>>(scores, logZ);
  nms_kernel   <<<B_IMG * 200, 256, 0, stream>>>(scores, logZ, tiles, maxconf);
  topk_kernel  <<<B_IMG, 512, 0, stream>>>(maxconf, topk);
  gather_kernel<<<B_IMG * TOPK_K, 256, 0, stream>>>(features, topk, out);
}